// VSDGATRNN_7249904795893
// MI455X (gfx1250) — compile-verified
//
#include <hip/hip_runtime.h>
#include <math.h>

// ---------------- problem constants ----------------
#define B_   32
#define T_   64
#define N_   128
#define D_   64
#define C_   129
#define CP_  132      // C padded to multiple of 4 (WMMA K step)
#define KP_  66       // CP_/2 : K-pair count for pair-interleaved weights
#define CPW_ 144      // comb / vW column pad (9 x 16 col-tiles, guard-free B loads)
#define CK_  16
#define QD_  5
#define NEGV (-9.0e15f)
#define NT_  512      // 16 wave32s per block

typedef float v2f __attribute__((ext_vector_type(2)));
typedef float v8f __attribute__((ext_vector_type(8)));

// ---------------- LDS layout (floats) ----------------
#define OFF_H    0                 // [128][64]
#define OFF_COMB 8192              // [128][144]  x | rar | h | 0-pad
#define OFF_S    26624             // [128][128]  (reused: r-gate, u-gate)
#define OFF_PC   43008             // [128][132]  (reused: Q[128][16]+K[128][16] in B1/B2; cand)
#define OFF_ATT  59904             // [128][132]
#define OFF_RAR  76800             // [128]
#define OFF_MASK 76928             // [128]
#define OFF_QV   77056             // [128][5]
#define OFF_VTO  77696             // [128]
#define SM_FLOATS 77824            // 311296 bytes < 320KB WGP LDS

// ---------------- workspace layout (floats) ----------------
// Weights stored K-pair interleaved: W2[((kp)*ld + n)*2 + j] = W[2*kp+j][n]
#define WS_QV   0                  // [128*5]
#define WS_NE   640                // [128*8]
#define WS_ADJ  1664               // [128*128]
#define WS_QWP  18048              // [2][66][16][2]
#define WS_KWP  22272              // [2][66][16][2]
#define WS_VWP  26496              // [2][66][144][2]
#define WS_VBP  64512              // [2][144]
#define WS_RWP  64800              // [5][66][64][2]
#define WS_UWP  107040             // [5][66][64][2]
#define WS_CWP  149280             // [5][66][64][2]
#define WS_TOT  191520             // 766080 bytes

static __device__ __forceinline__ v2f ld2(const float* p) {
  return *(const v2f*)p;           // 8-byte aligned pair load (b64)
}

// One 16x16 output tile per wave; K-loop over V_WMMA_F32_16X16X4_F32.
// Per ISA layouts (32-bit A 16x4 / B 4x16 / C,D 16x16):
//   lanes 0-15 hold K = kb,kb+1 ; lanes 16-31 hold K = kb+2,kb+3
//   D element r: row = r + 8*(lane>=16), col = lane&15
// Loaders return the (k0,k0+1) pair as one v2f; must be guard-free.
template <typename FA2, typename FB2>
static __device__ __forceinline__ v8f wmma_kloop(int K, v8f acc, FA2 la2, FB2 lb2) {
  const int lane = threadIdx.x & 31;
  const int hf   = lane >> 4;
  const int l15  = lane & 15;
  for (int kb = 0; kb < K; kb += 4) {
    const int k0 = kb + 2 * hf;
    const v2f a = la2(l15, k0);
    const v2f b = lb2(k0, l15);
    acc = __builtin_amdgcn_wmma_f32_16x16x4_f32(false, a, false, b, (short)0, acc,
                                                false, false);
  }
  return acc;
}

// ---------------- precompute kernels ----------------
// Per-node query vectors qv[N,5] and normalized node embeddings ne[N,8].
__global__ __launch_bounds__(128) void pre_embed(
    const float* __restrict__ plm,
    const float* __restrict__ pfW1, const float* __restrict__ pfb1,
    const float* __restrict__ pfW2, const float* __restrict__ pfb2,
    const float* __restrict__ pgW1, const float* __restrict__ pgb1,
    const float* __restrict__ pgW2, const float* __restrict__ pgb2,
    float* __restrict__ qv_g, float* __restrict__ ne_g) {
  __shared__ float hid[128];
  __shared__ float outv[8];
  const int n = blockIdx.x, j = threadIdx.x;
  const float* row = plm + (size_t)n * 768;
  float a = 0.f;
  for (int c = 0; c < 768; ++c) a += row[c] * pfW1[c * 128 + j];
  hid[j] = fmaxf(a + pfb1[j], 0.f);
  __syncthreads();
  if (j < QD_) {
    float s = 0.f;
    for (int i = 0; i < 128; ++i) s += hid[i] * pfW2[i * QD_ + j];
    qv_g[n * QD_ + j] = s + pfb2[j];
  }
  __syncthreads();
  float a2 = 0.f;
  for (int c = 0; c < 768; ++c) a2 += row[c] * pgW1[c * 128 + j];
  hid[j] = fmaxf(a2 + pgb1[j], 0.f);
  __syncthreads();
  if (j < 8) {
    float s = 0.f;
    for (int i = 0; i < 128; ++i) s += hid[i] * pgW2[i * 8 + j];
    outv[j] = s + pgb2[j];
  }
  __syncthreads();
  if (j == 0) {
    float nrm = 0.f;
    for (int e = 0; e < 8; ++e) nrm += outv[e] * outv[e];
    nrm = fmaxf(sqrtf(nrm), 1e-12f);
    for (int e = 0; e < 8; ++e) ne_g[n * 8 + e] = outv[e] / nrm;
  }
}

// adj = softmax(ne @ ne^T, axis=-1)  : one block per row
__global__ __launch_bounds__(128) void pre_adj(const float* __restrict__ ne_g,
                                               float* __restrict__ adj_g) {
  __shared__ float red[128];
  const int n = blockIdx.x, m = threadIdx.x;
  float s = 0.f;
  for (int e = 0; e < 8; ++e) s += ne_g[n * 8 + e] * ne_g[m * 8 + e];
  red[m] = s;
  __syncthreads();
  for (int off = 64; off > 0; off >>= 1) {
    if (m < off) red[m] = fmaxf(red[m], red[m + off]);
    __syncthreads();
  }
  const float mx = red[0];
  __syncthreads();
  const float e = expf(s - mx);
  red[m] = e;
  __syncthreads();
  for (int off = 64; off > 0; off >>= 1) {
    if (m < off) red[m] += red[m + off];
    __syncthreads();
  }
  adj_g[n * 128 + m] = e / red[0];
}

// Zero-padded, K-pair-interleaved weight copies: every WMMA B-fragment
// becomes a single aligned global_load_b64 with no guards.
__global__ __launch_bounds__(256) void pre_pad(
    const float* __restrict__ qW, const float* __restrict__ kW,
    const float* __restrict__ vW, const float* __restrict__ vb,
    const float* __restrict__ rstW, const float* __restrict__ updW,
    const float* __restrict__ candW,
    float* __restrict__ qWp, float* __restrict__ kWp,
    float* __restrict__ vWp, float* __restrict__ vbp,
    float* __restrict__ rWp, float* __restrict__ uWp, float* __restrict__ cWp) {
  const int g = blockIdx.x * blockDim.x + threadIdx.x;
  const int gs = gridDim.x * blockDim.x;
  for (int i = g; i < 2 * CP_ * CK_; i += gs) {   // iterate (h,k,n); q/k pair layout
    const int n = i % CK_, k = (i / CK_) % CP_, h = i / (CK_ * CP_);
    const int dst = ((h * KP_ + (k >> 1)) * CK_ + n) * 2 + (k & 1);
    const int src = (h * C_ + k) * CK_ + n;
    qWp[dst] = (k < C_) ? qW[src] : 0.f;
    kWp[dst] = (k < C_) ? kW[src] : 0.f;
  }
  for (int i = g; i < 2 * CP_ * CPW_; i += gs) {  // v pair layout [2][66][144][2]
    const int n = i % CPW_, k = (i / CPW_) % CP_, h = i / (CPW_ * CP_);
    const int dst = ((h * KP_ + (k >> 1)) * CPW_ + n) * 2 + (k & 1);
    vWp[dst] = (k < C_ && n < C_) ? vW[(h * C_ + k) * C_ + n] : 0.f;
  }
  for (int i = g; i < 2 * CPW_; i += gs) {        // vb: [2][144]
    const int n = i % CPW_, h = i / CPW_;
    vbp[i] = (n < C_) ? vb[h * C_ + n] : 0.f;
  }
  for (int i = g; i < QD_ * CP_ * D_; i += gs) {  // gates pair layout [5][66][64][2]
    const int n = i & 63, k = (i >> 6) % CP_, q = i / (CP_ * D_);
    const int dst = ((q * KP_ + (k >> 1)) * D_ + n) * 2 + (k & 1);
    const int src = (q * C_ + k) * D_ + n;
    const bool ok = (k < C_);
    rWp[dst] = ok ? rstW[src] : 0.f;
    uWp[dst] = ok ? updW[src] : 0.f;
    cWp[dst] = ok ? candW[src] : 0.f;
  }
}

// ---------------- main persistent recurrent kernel ----------------
// grid = B (one workgroup per batch element), block = 512 threads = 16 waves.
__global__ __launch_bounds__(NT_, 1) void vsd_main(
    const float* __restrict__ obs_emb, const float* __restrict__ obsmask,
    const float* __restrict__ avgint, const int* __restrict__ lengths,
    const float* __restrict__ rarW,
    const float* __restrict__ rWp, const float* __restrict__ rstb,
    const float* __restrict__ uWp, const float* __restrict__ updb,
    const float* __restrict__ cWp, const float* __restrict__ candb,
    const float* __restrict__ qWp, const float* __restrict__ qb,
    const float* __restrict__ kWp, const float* __restrict__ kb,
    const float* __restrict__ vWp, const float* __restrict__ vbp,
    const float* __restrict__ qv_g, const float* __restrict__ adj_g,
    float* __restrict__ out) {
  extern __shared__ float sm[];
  float* Hs    = sm + OFF_H;
  float* COMBs = sm + OFF_COMB;       // stride CPW_ (144)
  float* Ss    = sm + OFF_S;          // stride N_ (128)
  float* Rs    = sm + OFF_S;          // reuse after attention
  float* Us    = sm + OFF_S + 8192;
  float* Qs    = sm + OFF_PC;         // [128][16]  (B1/B2 only)
  float* Ks    = sm + OFF_PC + 2048;  // [128][16]  (B1/B2 only)
  float* PCs   = sm + OFF_PC;         // stride CP_ (132), B4/B5
  float* CANDs = sm + OFF_PC;         // reuse in C3/C4
  float* ATTs  = sm + OFF_ATT;        // stride CP_ (132)
  float* RARs  = sm + OFF_RAR;
  float* MASKs = sm + OFF_MASK;
  float* QVs   = sm + OFF_QV;
  float* VTOs  = sm + OFF_VTO;

  const int b    = blockIdx.x;
  const int tid  = threadIdx.x;
  const int w    = tid >> 5;
  const int lane = tid & 31;
  const int hf   = lane >> 4;
  const int l15  = lane & 15;

  // ---- per-dispatch init ----
  for (int i = tid; i < N_ * QD_; i += NT_) QVs[i] = qv_g[i];
  for (int n = tid; n < N_; n += NT_) {
    float s = 0.f;
    for (int t = 0; t < T_; ++t) s += obsmask[((size_t)b * T_ + t) * N_ + n];
    VTOs[n] = s;
  }
  for (int i = tid; i < N_ * D_; i += NT_) Hs[i] = 0.f;
  __syncthreads();
  const int len = lengths[b];

  for (int t = 0; t < T_; ++t) {
    const size_t bt = (size_t)b * T_ + t;
    const float* obs_t = obs_emb + bt * N_ * D_;

    // ---- Phase A: rarity, mask, comb = [x | rar | h | 0], zero attacc ----
    for (int n = tid; n < N_; n += NT_) {
      MASKs[n] = obsmask[bt * N_ + n];
      RARs[n]  = 0.5f * tanhf(avgint[bt * N_ + n] / (VTOs[n] + 1.0f));
    }
    __syncthreads();
    for (int i = tid; i < N_ * CPW_; i += NT_) {
      const int n = i / CPW_, c = i - n * CPW_;
      float v;
      if (c < D_)        v = obs_t[n * D_ + c];
      else if (c == D_)  v = RARs[n];
      else if (c < C_)   v = Hs[n * D_ + (c - D_ - 1)];
      else               v = 0.f;
      COMBs[i] = v;
    }
    for (int i = tid; i < N_ * CP_; i += NT_) ATTs[i] = 0.f;
    __syncthreads();

    for (int head = 0; head < 2; ++head) {
      // ---- B1: q = comb@qW+qb, k = comb@kW+kb  (16 tiles -> 16 waves) ----
      {
        const int isK = w >> 3, tm = w & 7;
        const float* W  = isK ? kWp : qWp;
        const float* Bv = isK ? kb : qb;
        float* dst = isK ? Ks : Qs;
        v8f acc = {};
        acc = wmma_kloop(CP_, acc,
            [&](int m, int k0) { return ld2(&COMBs[(tm * 16 + m) * CPW_ + k0]); },
            [&](int k0, int n) {
              return ld2(&W[((head * KP_ + (k0 >> 1)) * CK_ + n) * 2]);
            });
#pragma unroll
        for (int r = 0; r < 8; ++r) {
          const int m = tm * 16 + r + 8 * hf;
          dst[m * CK_ + l15] = acc[r] + Bv[head * CK_ + l15];
        }
      }
      __syncthreads();

      // ---- B2: s = leaky(q k^T * 0.25), adjacency mask (64 tiles) ----
      for (int tt = w; tt < 64; tt += 16) {
        const int tm = tt >> 3, tn = tt & 7;
        v8f acc = {};
        acc = wmma_kloop(CK_, acc,
            [&](int m, int k0) { return ld2(&Qs[(tm * 16 + m) * CK_ + k0]); },
            [&](int k0, int n) { return ld2(&Ks[(tn * 16 + n) * CK_ + k0]); });
#pragma unroll
        for (int r = 0; r < 8; ++r) {
          const int m = tm * 16 + r + 8 * hf;
          const int n = tn * 16 + l15;
          float v = acc[r] * 0.25f;
          v = (v > 0.f) ? v : 0.2f * v;
          if (m != n) {
            const float a = adj_g[m * N_ + n] *
                            (1.0f - rarW[m * N_ + n] * fabsf(RARs[m] - RARs[n])) *
                            MASKs[m] * MASKs[n];
            if (a == 0.f) v = NEGV;
          }
          Ss[m * N_ + n] = v;
        }
      }
      __syncthreads();

      // ---- B3: row softmax, fold 0.5 head-mean (8 rows per wave) ----
      for (int row = w; row < N_; row += 16) {
        float x0 = Ss[row * N_ + lane], x1 = Ss[row * N_ + lane + 32];
        float x2 = Ss[row * N_ + lane + 64], x3 = Ss[row * N_ + lane + 96];
        float mx = fmaxf(fmaxf(x0, x1), fmaxf(x2, x3));
        for (int off = 16; off > 0; off >>= 1) mx = fmaxf(mx, __shfl_xor(mx, off, 32));
        float e0 = expf(x0 - mx), e1 = expf(x1 - mx);
        float e2 = expf(x2 - mx), e3 = expf(x3 - mx);
        float sum = e0 + e1 + e2 + e3;
        for (int off = 16; off > 0; off >>= 1) sum += __shfl_xor(sum, off, 32);
        const float inv = 0.5f / sum;     // 0.5 = mean over 2 heads
        Ss[row * N_ + lane]      = e0 * inv;
        Ss[row * N_ + lane + 32] = e1 * inv;
        Ss[row * N_ + lane + 64] = e2 * inv;
        Ss[row * N_ + lane + 96] = e3 * inv;
      }
      __syncthreads();

      // ---- B4: pc = p @ comb   (8x9 = 72 tiles) ----
      for (int tt = w; tt < 72; tt += 16) {
        const int tm = tt / 9, tn = tt % 9;
        v8f acc = {};
        acc = wmma_kloop(N_, acc,
            [&](int m, int k0) { return ld2(&Ss[(tm * 16 + m) * N_ + k0]); },
            [&](int k0, int n) {
              v2f bb;
              bb.x = COMBs[k0 * CPW_ + tn * 16 + n];
              bb.y = COMBs[(k0 + 1) * CPW_ + tn * 16 + n];
              return bb;
            });
#pragma unroll
        for (int r = 0; r < 8; ++r) {
          const int m = tm * 16 + r + 8 * hf;
          const int nn = tn * 16 + l15;
          if (nn < CP_) PCs[m * CP_ + nn] = acc[r];
        }
      }
      __syncthreads();

      // ---- B5: attacc += pc @ vW[head] + 0.5*vb[head]  (72 tiles) ----
      for (int tt = w; tt < 72; tt += 16) {
        const int tm = tt / 9, tn = tt % 9;
        v8f acc;
#pragma unroll
        for (int r = 0; r < 8; ++r) {
          const int m = tm * 16 + r + 8 * hf;
          const int nn = tn * 16 + l15;
          acc[r] = (nn < CP_) ? ATTs[m * CP_ + nn] : 0.f;
        }
        acc = wmma_kloop(CP_, acc,
            [&](int m, int k0) { return ld2(&PCs[(tm * 16 + m) * CP_ + k0]); },
            [&](int k0, int n) {
              return ld2(&vWp[((head * KP_ + (k0 >> 1)) * CPW_ + tn * 16 + n) * 2]);
            });
#pragma unroll
        for (int r = 0; r < 8; ++r) {
          const int m = tm * 16 + r + 8 * hf;
          const int nn = tn * 16 + l15;
          if (nn < CP_) ATTs[m * CP_ + nn] = acc[r] + 0.5f * vbp[head * CPW_ + nn];
        }
      }
      __syncthreads();
    }  // head

    // ---- C1: r/u gates: sigmoid(sum_q (qv_q ⊙ att) @ W_q + qv@b)  (64 tiles) ----
    for (int tt = w; tt < 64; tt += 16) {
      const int g = tt >> 5, tile = tt & 31;
      const int tm = tile >> 2, tn = tile & 3;
      const float* W  = g ? uWp : rWp;
      const float* bb = g ? updb : rstb;
      float* dst = g ? Us : Rs;
      v8f acc = {};
      for (int q = 0; q < QD_; ++q) {
        acc = wmma_kloop(CP_, acc,
            [&](int m, int k0) {
              const int mm = tm * 16 + m;
              v2f a = ld2(&ATTs[mm * CP_ + k0]);
              const float sc = QVs[mm * QD_ + q];
              a.x *= sc; a.y *= sc;
              return a;
            },
            [&](int k0, int n) {
              return ld2(&W[((q * KP_ + (k0 >> 1)) * D_ + tn * 16 + n) * 2]);
            });
      }
#pragma unroll
      for (int r = 0; r < 8; ++r) {
        const int m = tm * 16 + r + 8 * hf;
        const int n = tn * 16 + l15;
        float bias = 0.f;
        for (int q = 0; q < QD_; ++q) bias += QVs[m * QD_ + q] * bb[q * D_ + n];
        const float x = acc[r] + bias;
        dst[m * D_ + n] = 1.f / (1.f + expf(-x));
      }
    }
    __syncthreads();

    // ---- C2: h1 = observed ? r*h : h ; write into h and into comb h-slot ----
    for (int i = tid; i < N_ * D_; i += NT_) {
      const int n = i >> 6, c = i & 63;
      const float hv = Hs[i];
      const float h1 = (MASKs[n] > 0.f) ? Rs[i] * hv : hv;
      Hs[i] = h1;
      COMBs[n * CPW_ + D_ + 1 + c] = h1;
    }
    __syncthreads();

    // ---- C3: cand = tanh(gate(concat[x,h1]))  (32 tiles) ----
    for (int tt = w; tt < 32; tt += 16) {
      const int tm = tt >> 2, tn = tt & 3;
      v8f acc = {};
      for (int q = 0; q < QD_; ++q) {
        acc = wmma_kloop(CP_, acc,
            [&](int m, int k0) {
              const int mm = tm * 16 + m;
              v2f a = ld2(&COMBs[mm * CPW_ + k0]);
              const float sc = QVs[mm * QD_ + q];
              a.x *= sc; a.y *= sc;
              return a;
            },
            [&](int k0, int n) {
              return ld2(&cWp[((q * KP_ + (k0 >> 1)) * D_ + tn * 16 + n) * 2]);
            });
      }
#pragma unroll
      for (int r = 0; r < 8; ++r) {
        const int m = tm * 16 + r + 8 * hf;
        const int n = tn * 16 + l15;
        float bias = 0.f;
        for (int q = 0; q < QD_; ++q) bias += QVs[m * QD_ + q] * candb[q * D_ + n];
        CANDs[m * D_ + n] = tanhf(acc[r] + bias);
      }
    }
    __syncthreads();

    // ---- C4: h2 = observed ? (1-u)*h1 + u*cand : h1 ; capture at t==len-1 ----
    for (int i = tid; i < N_ * D_; i += NT_) {
      const int n = i >> 6, c = i & 63;
      const float h1 = Hs[i];
      float h2 = h1;
      if (MASKs[n] > 0.f) {
        const float u = Us[i];
        h2 = (1.f - u) * h1 + u * CANDs[i];
      }
      Hs[i] = h2;
      if (t == len - 1) out[((size_t)b * N_ + n) * D_ + c] = h2;
    }
    __syncthreads();
  }  // t
}

// ---------------- host entry ----------------
extern "C" void kernel_launch(void* const* d_in, const int* in_sizes, int n_in,
                              void* d_out, int out_size, void* d_ws, size_t ws_size,
                              hipStream_t stream) {
  const float* obs_emb       = (const float*)d_in[0];
  const float* observed_mask = (const float*)d_in[1];
  const float* avg_interval  = (const float*)d_in[2];
  const float* plm           = (const float*)d_in[3];
  const int*   lengths       = (const int*)d_in[4];
  const float* rarity_W      = (const float*)d_in[5];
  const float* pf_W1 = (const float*)d_in[6];
  const float* pf_b1 = (const float*)d_in[7];
  const float* pf_W2 = (const float*)d_in[8];
  const float* pf_b2 = (const float*)d_in[9];
  const float* pg_W1 = (const float*)d_in[10];
  const float* pg_b1 = (const float*)d_in[11];
  const float* pg_W2 = (const float*)d_in[12];
  const float* pg_b2 = (const float*)d_in[13];
  const float* rst_W = (const float*)d_in[14];
  const float* rst_b = (const float*)d_in[15];
  const float* upd_W = (const float*)d_in[16];
  const float* upd_b = (const float*)d_in[17];
  const float* cand_W = (const float*)d_in[18];
  const float* cand_b = (const float*)d_in[19];
  const float* q_W = (const float*)d_in[20];
  const float* q_b = (const float*)d_in[21];
  const float* k_W = (const float*)d_in[22];
  const float* k_b = (const float*)d_in[23];
  const float* v_W = (const float*)d_in[24];
  const float* v_b = (const float*)d_in[25];
  float* out = (float*)d_out;

  float* ws    = (float*)d_ws;
  float* qv_g  = ws + WS_QV;
  float* ne_g  = ws + WS_NE;
  float* adj_g = ws + WS_ADJ;
  float* qWp   = ws + WS_QWP;
  float* kWp   = ws + WS_KWP;
  float* vWp   = ws + WS_VWP;
  float* vbp   = ws + WS_VBP;
  float* rWp   = ws + WS_RWP;
  float* uWp   = ws + WS_UWP;
  float* cWp   = ws + WS_CWP;

  pre_embed<<<N_, 128, 0, stream>>>(plm, pf_W1, pf_b1, pf_W2, pf_b2,
                                    pg_W1, pg_b1, pg_W2, pg_b2, qv_g, ne_g);
  pre_adj<<<N_, 128, 0, stream>>>(ne_g, adj_g);
  pre_pad<<<64, 256, 0, stream>>>(q_W, k_W, v_W, v_b, rst_W, upd_W, cand_W,
                                  qWp, kWp, vWp, vbp, rWp, uWp, cWp);

  (void)hipFuncSetAttribute((const void*)vsd_main,
                            hipFuncAttributeMaxDynamicSharedMemorySize,
                            SM_FLOATS * (int)sizeof(float));
  vsd_main<<<B_, NT_, SM_FLOATS * sizeof(float), stream>>>(
      obs_emb, observed_mask, avg_interval, lengths, rarity_W,
      rWp, rst_b, uWp, upd_b, cWp, cand_b,
      qWp, q_b, kWp, k_b, vWp, vbp, qv_g, adj_g, out);
}